// GPSNet_34067680592555
// MI455X (gfx1250) — compile-verified
//
#include <hip/hip_runtime.h>
#include <hip/hip_bf16.h>

// ---------------------------------------------------------------------------
// GAT layer:  h = x@W ; logits ; edge segment-softmax ; scatter-aggregate
// N=100000, E=1600000, HEADS=4, OUT_C=32, IN_C=128 (HEADS*OUT_C = 128)
// ---------------------------------------------------------------------------

typedef float v2f __attribute__((ext_vector_type(2)));
typedef float v8f __attribute__((ext_vector_type(8)));

#define IN_C      128
#define HC        128   // HEADS * OUT_C
#define HEADS     4
#define OUT_C     32
#define NEG_SLOPE 0.2f
#define EPS_F     1e-16f

#define GEMM_ROWS 64            // rows of x per block
#define KPANEL    32            // K panel staged in LDS
#define SW_STRIDE 136           // W panel row stride (floats): 544B rows, 16B-aligned,
                                //   lanes 0-15 (row k) and 16-31 (row k+2) hit disjoint banks
#define SX_STRIDE 36            // x panel row stride (floats): 144B rows, 16B-aligned,
                                //   m*36 mod 64 is a permutation -> conflict-free A reads

// ---------------------------------------------------------------------------
// gfx1250 async global->LDS copy (ASYNCcnt-tracked, bypasses VGPR round-trip)
// ---------------------------------------------------------------------------
typedef __attribute__((address_space(3))) char lds_char;

__device__ __forceinline__ unsigned lds_byte_off(const void* p) {
  // generic -> LDS addrspacecast, then ptrtoint: wave-relative LDS byte address
  return (unsigned)(unsigned long long)(lds_char*)(void*)p;
}

__device__ __forceinline__ void async_g2l_b128(unsigned lds_off, const void* gptr) {
  asm volatile("global_load_async_to_lds_b128 %0, %1, off"
               :: "v"(lds_off), "v"((unsigned long long)gptr)
               : "memory");
}

__device__ __forceinline__ void wait_async0() {
  asm volatile("s_wait_asynccnt 0" ::: "memory");
}

// ---------------------------------------------------------------------------
// K0: init out = bias (broadcast), m = -inf, s = 0
// ---------------------------------------------------------------------------
__global__ __launch_bounds__(256) void gat_init(float* __restrict__ out,
                                                const float* __restrict__ bias,
                                                float* __restrict__ mmax,
                                                float* __restrict__ ssum,
                                                int n) {
  int i = blockIdx.x * 256 + threadIdx.x;
  if (i < n * HC) out[i] = bias[i & (HC - 1)];
  if (i < n * HEADS) {
    mmax[i] = -__builtin_huge_valf();
    ssum[i] = 0.0f;
  }
}

// ---------------------------------------------------------------------------
// K1: h = x @ W   via V_WMMA_F32_16X16X4_F32 (fp32 in/out, wave32)
//   block: 256 threads = 8 waves; 64 rows x 128 cols of output.
//   wave w owns col-tile [16w,16w+16) for 4 row-tiles of 16.
//   Panels staged with GLOBAL_LOAD_ASYNC_TO_LDS_B128 + s_wait_asynccnt.
// ---------------------------------------------------------------------------
__global__ __launch_bounds__(256) void gat_gemm(const float* __restrict__ x,
                                                const float* __restrict__ W,
                                                float* __restrict__ h,
                                                int n) {
  __shared__ __align__(16) float sW[KPANEL * SW_STRIDE];     // 32 x 128 W panel
  __shared__ __align__(16) float sX[GEMM_ROWS * SX_STRIDE];  // 64 x 32 x panel

  const int tid  = threadIdx.x;
  const int row0 = blockIdx.x * GEMM_ROWS;

  const int wave = tid >> 5;          // 0..7 -> column tile
  const int lane = tid & 31;
  const int m    = lane & 15;         // A row within tile / D col
  const int kb   = (lane >> 4) << 1;  // 0 or 2: K sub-offset for A/B frags
  const int colt = wave << 4;         // output column base for this wave

  const unsigned sW0 = lds_byte_off(sW);
  const unsigned sX0 = lds_byte_off(sX);

  v8f acc[4];
  acc[0] = {}; acc[1] = {}; acc[2] = {}; acc[3] = {};

  for (int kp = 0; kp < IN_C; kp += KPANEL) {
    // ---- async-stage W panel: rows kp..kp+31, all 128 cols (b128/lane) ----
    // 1024 float4 / 256 threads = 4 issues, full EXEC
    for (int i = tid; i < KPANEL * 32; i += 256) {
      int r  = i >> 5;                // 32 float4 per W row
      int c4 = i & 31;
      async_g2l_b128(sW0 + (unsigned)(r * SW_STRIDE + c4 * 4) * 4u,
                     W + (kp + r) * HC + c4 * 4);
    }
    // ---- async-stage x panel: 64 rows, cols kp..kp+31 (b128/lane) ----
    // 512 float4 / 256 threads = 2 issues; clamp OOB rows (their D rows are
    // never stored, so garbage there is harmless) to keep EXEC all-ones.
    for (int i = tid; i < GEMM_ROWS * 8; i += 256) {
      int r  = i >> 3;                // 8 float4 per staged row
      int c4 = i & 7;
      int gr = row0 + r;
      if (gr >= n) gr = n - 1;
      async_g2l_b128(sX0 + (unsigned)(r * SX_STRIDE + c4 * 4) * 4u,
                     x + (long)gr * IN_C + kp + c4 * 4);
    }
    wait_async0();
    __syncthreads();

    // ---- 8 x (K=4) WMMA steps over this panel ----
#pragma unroll
    for (int kk = 0; kk < KPANEL; kk += 4) {
      v2f b;
      b.x = sW[(kk + kb)     * SW_STRIDE + colt + m];
      b.y = sW[(kk + kb + 1) * SW_STRIDE + colt + m];
#pragma unroll
      for (int rt = 0; rt < 4; ++rt) {
        const float* xr = &sX[(rt * 16 + m) * SX_STRIDE + kk + kb];
        v2f a;
        a.x = xr[0];
        a.y = xr[1];
        acc[rt] = __builtin_amdgcn_wmma_f32_16x16x4_f32(
            /*neg_a=*/false, a, /*neg_b=*/false, b,
            /*c_mod=*/(short)0, acc[rt], /*reuse_a=*/false, /*reuse_b=*/false);
      }
    }
    __syncthreads();
  }

  // ---- store D tiles: lane l, vgpr v -> row v + 8*(l>=16), col l&15 ----
  const int rhalf = (lane >> 4) << 3;
#pragma unroll
  for (int rt = 0; rt < 4; ++rt) {
    int rbase = row0 + rt * 16 + rhalf;
#pragma unroll
    for (int v = 0; v < 8; ++v) {
      int r = rbase + v;
      if (r < n) h[(long)r * HC + colt + m] = acc[rt][v];
    }
  }
}

// ---------------------------------------------------------------------------
// K2: per-node logits  a_src[n,h] = <h[n,h,:], att_src[h,:]> (wave per pair)
// ---------------------------------------------------------------------------
__global__ __launch_bounds__(256) void gat_logits(const float* __restrict__ h,
                                                  const float* __restrict__ att_src,
                                                  const float* __restrict__ att_dst,
                                                  float* __restrict__ a_src,
                                                  float* __restrict__ a_dst,
                                                  int n) {
  int gw   = (blockIdx.x * 256 + threadIdx.x) >> 5;
  int lane = threadIdx.x & 31;
  int node = gw >> 2;
  int head = gw & 3;
  if (node >= n) return;

  float v  = h[(long)node * HC + head * OUT_C + lane];
  float ws = v * att_src[head * OUT_C + lane];
  float wd = v * att_dst[head * OUT_C + lane];
#pragma unroll
  for (int off = 16; off > 0; off >>= 1) {
    ws += __shfl_xor(ws, off, 32);
    wd += __shfl_xor(wd, off, 32);
  }
  if (lane == 0) {
    a_src[node * HEADS + head] = ws;
    a_dst[node * HEADS + head] = wd;
  }
}

// ---------------------------------------------------------------------------
// float atomic-max via monotone int punning
// ---------------------------------------------------------------------------
__device__ __forceinline__ void atomicMaxFloat(float* addr, float val) {
  if (val >= 0.0f)
    atomicMax((int*)addr, __float_as_int(val));
  else
    atomicMin((unsigned int*)addr, (unsigned int)__float_as_int(val));
}

__device__ __forceinline__ void edge_nodes(int e, int E_, const int* __restrict__ ei,
                                           int& s, int& d) {
  if (e < E_) { s = ei[e]; d = ei[E_ + e]; }
  else        { s = d = e - E_; }          // self loop
}

__device__ __forceinline__ float leaky(float v) {
  return v >= 0.0f ? v : NEG_SLOPE * v;
}

// ---------------------------------------------------------------------------
// K3: segment max of leaky-relu edge scores (one thread per edge)
// ---------------------------------------------------------------------------
__global__ __launch_bounds__(256) void gat_edge_max(const int* __restrict__ ei,
                                                    const float* __restrict__ a_src,
                                                    const float* __restrict__ a_dst,
                                                    float* __restrict__ mmax,
                                                    int E_, int n) {
  int e = blockIdx.x * 256 + threadIdx.x;
  if (e >= E_ + n) return;
  int s, d;
  edge_nodes(e, E_, ei, s, d);
  float4 as = ((const float4*)a_src)[s];
  float4 ad = ((const float4*)a_dst)[d];
  atomicMaxFloat(&mmax[d * HEADS + 0], leaky(as.x + ad.x));
  atomicMaxFloat(&mmax[d * HEADS + 1], leaky(as.y + ad.y));
  atomicMaxFloat(&mmax[d * HEADS + 2], leaky(as.z + ad.z));
  atomicMaxFloat(&mmax[d * HEADS + 3], leaky(as.w + ad.w));
}

// ---------------------------------------------------------------------------
// K4: segment sum of exp(e - m[dst])
// ---------------------------------------------------------------------------
__global__ __launch_bounds__(256) void gat_edge_sum(const int* __restrict__ ei,
                                                    const float* __restrict__ a_src,
                                                    const float* __restrict__ a_dst,
                                                    const float* __restrict__ mmax,
                                                    float* __restrict__ ssum,
                                                    int E_, int n) {
  int e = blockIdx.x * 256 + threadIdx.x;
  if (e >= E_ + n) return;
  int s, d;
  edge_nodes(e, E_, ei, s, d);
  float4 as = ((const float4*)a_src)[s];
  float4 ad = ((const float4*)a_dst)[d];
  float4 mm = ((const float4*)mmax)[d];
  atomicAdd(&ssum[d * HEADS + 0], __expf(leaky(as.x + ad.x) - mm.x));
  atomicAdd(&ssum[d * HEADS + 1], __expf(leaky(as.y + ad.y) - mm.y));
  atomicAdd(&ssum[d * HEADS + 2], __expf(leaky(as.z + ad.z) - mm.z));
  atomicAdd(&ssum[d * HEADS + 3], __expf(leaky(as.w + ad.w) - mm.w));
}

// ---------------------------------------------------------------------------
// K5: out[dst] += alpha * h[src]   (one wave32 per edge; lane = channel)
// ---------------------------------------------------------------------------
__global__ __launch_bounds__(256) void gat_aggregate(const int* __restrict__ ei,
                                                     const float* __restrict__ a_src,
                                                     const float* __restrict__ a_dst,
                                                     const float* __restrict__ mmax,
                                                     const float* __restrict__ ssum,
                                                     const float* __restrict__ h,
                                                     float* __restrict__ out,
                                                     int E_, int n) {
  int gw   = (blockIdx.x * 256 + threadIdx.x) >> 5;
  int lane = threadIdx.x & 31;
  if (gw >= E_ + n) return;
  int s, d;
  edge_nodes(gw, E_, ei, s, d);

  float4 as = ((const float4*)a_src)[s];
  float4 ad = ((const float4*)a_dst)[d];
  float4 mm = ((const float4*)mmax)[d];
  float4 sm = ((const float4*)ssum)[d];

  float alpha[HEADS];
  alpha[0] = __expf(leaky(as.x + ad.x) - mm.x) / (sm.x + EPS_F);
  alpha[1] = __expf(leaky(as.y + ad.y) - mm.y) / (sm.y + EPS_F);
  alpha[2] = __expf(leaky(as.z + ad.z) - mm.z) / (sm.z + EPS_F);
  alpha[3] = __expf(leaky(as.w + ad.w) - mm.w) / (sm.w + EPS_F);

  const float* hs = h   + (long)s * HC;
  float*       od = out + (long)d * HC;
  __builtin_prefetch(hs, 0, 0);   // global_prefetch_b8
#pragma unroll
  for (int hh = 0; hh < HEADS; ++hh) {
    atomicAdd(&od[hh * OUT_C + lane], alpha[hh] * hs[hh * OUT_C + lane]);
  }
}

// ---------------------------------------------------------------------------
// launch
// ---------------------------------------------------------------------------
extern "C" void kernel_launch(void* const* d_in, const int* in_sizes, int n_in,
                              void* d_out, int out_size, void* d_ws, size_t ws_size,
                              hipStream_t stream) {
  const float* x       = (const float*)d_in[0];
  const int*   ei      = (const int*)d_in[1];
  const float* W       = (const float*)d_in[2];
  const float* att_src = (const float*)d_in[3];
  const float* att_dst = (const float*)d_in[4];
  const float* bias    = (const float*)d_in[5];
  float*       out     = (float*)d_out;

  const int n  = in_sizes[0] / IN_C;   // nodes
  const int E_ = in_sizes[1] / 2;      // edges (without self loops)

  // workspace layout
  char* ws = (char*)d_ws;
  float* h     = (float*)ws;                       ws += (size_t)n * HC * sizeof(float);
  float* a_src = (float*)ws;                       ws += (size_t)n * HEADS * sizeof(float);
  float* a_dst = (float*)ws;                       ws += (size_t)n * HEADS * sizeof(float);
  float* mmax  = (float*)ws;                       ws += (size_t)n * HEADS * sizeof(float);
  float* ssum  = (float*)ws;

  auto cdiv = [](long a, long b) { return (int)((a + b - 1) / b); };

  const int tot_edges = E_ + n;

  gat_init<<<cdiv((long)n * HC, 256), 256, 0, stream>>>(out, bias, mmax, ssum, n);
  gat_gemm<<<cdiv(n, GEMM_ROWS), 256, 0, stream>>>(x, W, h, n);
  gat_logits<<<cdiv((long)n * HEADS * 32, 256), 256, 0, stream>>>(h, att_src, att_dst,
                                                                  a_src, a_dst, n);
  gat_edge_max<<<cdiv(tot_edges, 256), 256, 0, stream>>>(ei, a_src, a_dst, mmax, E_, n);
  gat_edge_sum<<<cdiv(tot_edges, 256), 256, 0, stream>>>(ei, a_src, a_dst, mmax, ssum, E_, n);
  gat_aggregate<<<cdiv((long)tot_edges * 32, 256), 256, 0, stream>>>(ei, a_src, a_dst, mmax,
                                                                     ssum, h, out, E_, n);
}